// Downsample1d_16741782519811
// MI455X (gfx1250) — compile-verified
//
#include <hip/hip_runtime.h>

// Downsample1d: depthwise conv, K=4 kernel [1/8,3/8,3/8,1/8], stride 2, reflect pad 1.
// Input  [16, 512, 8192] f32  -> Output [16, 512, 4096] f32.
// Pure HBM stream (403 MB total, 0.58 FLOP/byte -> ~17 us floor @ 23.3 TB/s).
// gfx1250 path: GVS-mode async global->LDS B128 copies (ASYNCcnt) with NT load
// hints, LDS serves the overlapping stride-2 windows, NT b128 stores out.

#define L_IN    8192
#define L_OUT   4096
#define THREADS 256

typedef float v4f __attribute__((ext_vector_type(4)));

__global__ __launch_bounds__(THREADS)
void downsample1d_kernel(const float* __restrict__ x, float* __restrict__ y)
{
    __shared__ __align__(16) float s[L_IN];

    const int row = blockIdx.x;          // one (b, c) row per workgroup
    const int tid = threadIdx.x;
    const float* gx = x + (size_t)row * L_IN;   // uniform -> SGPR pair

    // ---- Async copy global -> LDS, GVS mode ----
    // IOFFSET advances BOTH addresses: it is added to the LDS address and to SADDR + VADDR.
    // 8 instructions x (256 lanes x 16 B) = 32 KB row, single VGPR of address math.
    unsigned lds0 = (unsigned)(size_t)(&s[tid * 4]); // low 32 bits of flat LDS ptr = LDS byte address
    unsigned voff = tid * 16;                        // per-lane global byte offset
    asm volatile(
        "global_load_async_to_lds_b128 %0, %1, %2 offset:0x0 th:TH_LOAD_NT\n\t"
        "global_load_async_to_lds_b128 %0, %1, %2 offset:0x1000 th:TH_LOAD_NT\n\t"
        "global_load_async_to_lds_b128 %0, %1, %2 offset:0x2000 th:TH_LOAD_NT\n\t"
        "global_load_async_to_lds_b128 %0, %1, %2 offset:0x3000 th:TH_LOAD_NT\n\t"
        "global_load_async_to_lds_b128 %0, %1, %2 offset:0x4000 th:TH_LOAD_NT\n\t"
        "global_load_async_to_lds_b128 %0, %1, %2 offset:0x5000 th:TH_LOAD_NT\n\t"
        "global_load_async_to_lds_b128 %0, %1, %2 offset:0x6000 th:TH_LOAD_NT\n\t"
        "global_load_async_to_lds_b128 %0, %1, %2 offset:0x7000 th:TH_LOAD_NT"
        :: "v"(lds0), "v"(voff), "s"(gx)
        : "memory");
    asm volatile("s_wait_asynccnt 0" ::: "memory");  // this wave's copies complete
    __syncthreads();                                  // all waves' copies visible

    // ---- Each thread: 16 outputs from inputs [32*tid - 1 .. 32*tid + 32] ----
    const int base = tid * 32;
    float f[32];
    #pragma unroll
    for (int k = 0; k < 8; ++k) {
        v4f v = *(const v4f*)(&s[base + 4 * k]);     // aligned ds_load_b128
        f[4*k + 0] = v.x; f[4*k + 1] = v.y; f[4*k + 2] = v.z; f[4*k + 3] = v.w;
    }
    // reflect: index -1 -> x[1], index 8192 -> x[8190]
    const float xl = (tid == 0)           ? s[1]        : s[base - 1];
    const float xr = (tid == THREADS - 1) ? s[L_IN - 2] : s[base + 32];

    float o[16];
    #pragma unroll
    for (int j = 0; j < 16; ++j) {
        const float a = (j == 0)  ? xl : f[2*j - 1];
        const float b = f[2*j];
        const float c = f[2*j + 1];
        const float d = (j == 15) ? xr : f[2*j + 2];
        o[j] = 0.125f * (a + d) + 0.375f * (b + c);
    }

    // ---- 4 x b128 non-temporal coalesced stores ----
    float* gy = y + (size_t)row * L_OUT + tid * 16;
    #pragma unroll
    for (int g = 0; g < 4; ++g) {
        v4f v = { o[4*g + 0], o[4*g + 1], o[4*g + 2], o[4*g + 3] };
        __builtin_nontemporal_store(v, (v4f*)(gy + 4 * g));
    }
}

extern "C" void kernel_launch(void* const* d_in, const int* in_sizes, int n_in,
                              void* d_out, int out_size, void* d_ws, size_t ws_size,
                              hipStream_t stream)
{
    const float* x = (const float*)d_in[0];
    float*       y = (float*)d_out;

    const int rows = in_sizes[0] / L_IN;   // 16 * 512 = 8192
    downsample1d_kernel<<<dim3(rows), dim3(THREADS), 0, stream>>>(x, y);
}